// MultiPersDiffFixecCenters_3624952397857
// MI455X (gfx1250) — compile-verified
//
#include <hip/hip_runtime.h>
#include <stdint.h>

#define EPS_C      1.0e-4f
#define GRID_RES_C 0.01f
#define ATOL_C     0.01f
#define RTOL_C     1.0e-5f

#define S_MAX          6000
#define WAVES_PER_BLK  8
#define BLK_THREADS    256

// ---------------------------------------------------------------------------
// Kernel A: build filtration, interleaved SoA-of-pairs: filt[2j]={fx}, filt[2j+1]={fy},
// and copy bars0/bars1 into the output (they are part of the returned tuple).
// ---------------------------------------------------------------------------
__global__ void build_filtration_kernel(const float* __restrict__ f_v,
                                        const int*   __restrict__ edges,
                                        const float* __restrict__ bars0,
                                        const float* __restrict__ bars1,
                                        float* __restrict__ out,
                                        float* __restrict__ filt,   // {fx,fy} pairs
                                        int V, int E, int PR) {
    int t = blockIdx.x * blockDim.x + threadIdx.x;
    if (t < PR) {                       // bars0 -> out[0..PR), bars1 -> out[PR..2PR)
        out[t]      = bars0[t];
        out[PR + t] = bars1[t];
    }
    if (t < V) {                        // vertex part of filtration (plain copy)
        filt[2 * t]     = f_v[2 * t];
        filt[2 * t + 1] = f_v[2 * t + 1];
    }
    if (t < E) {                        // edge part: max over endpoints + EPS
        int a = edges[2 * t];
        int b = edges[2 * t + 1];
        float ax = f_v[2 * a],     bx = f_v[2 * b];
        float ay = f_v[2 * a + 1], by = f_v[2 * b + 1];
        filt[2 * (V + t)]     = fmaxf(ax, bx) + EPS_C;
        filt[2 * (V + t) + 1] = fmaxf(ay, by) + EPS_C;
    }
}

// ---------------------------------------------------------------------------
// Per-element evaluation: returns (vx, vy) and ORs the "any" flags.
// ---------------------------------------------------------------------------
__device__ __forceinline__ float2 eval_cell(float fxj, float fyj,
                                            float s, float px, float py,
                                            float psum, float lastx, float lasty,
                                            bool& aLx, bool& aUx,
                                            bool& aLy, bool& aUy) {
    const float tx = fmaf(RTOL_C, fabsf(fxj), ATOL_C);
    const float ty = fmaf(RTOL_C, fabsf(fyj), ATOL_C);
    const float dx = px - fxj;              // line_x - fx = k*s + (px - fx)
    const float dy = py - fyj;

    bool hx = false, hy = false;
#pragma unroll
    for (int k = -2; k <= 2; ++k) {
        hx |= fabsf(fmaf((float)k, s, dx)) <= tx;
        hy |= fabsf(fmaf((float)k, s, dy)) <= ty;
    }

    const float fsum = fxj + fyj;
    const bool below = fsum < psum;
    const bool above = fsum > psum;
    const bool condx = fyj <= lasty;
    const bool condy = fxj <= lastx;

    const bool lx = hx && condx && below;
    const bool ux = hx && condx && above;
    const bool ly = hy && condy && below;
    const bool uy = hy && condy && above;

    aLx |= lx; aUx |= ux; aLy |= ly; aUy |= uy;

    return make_float2((ux ? 1.0f : 0.0f) - (lx ? 1.0f : 0.0f),
                       (uy ? 1.0f : 0.0f) - (ly ? 1.0f : 0.0f));
}

// ---------------------------------------------------------------------------
// Kernel B: one wave (32 lanes) per sample row p; blockIdx.y = r, blockIdx.z
// = bar set. Filtration (48 KB) staged into LDS with CDNA5 async-to-LDS b128
// copies (ASYNCcnt), fenced with s_wait_asynccnt before the workgroup barrier.
// Each lane handles a pair of adjacent j: one ds_load_b128 in, one
// global_store_b128 out (512 B per wave-store instruction).
// ---------------------------------------------------------------------------
__global__ __launch_bounds__(BLK_THREADS)
void grad_matrices_kernel(const float* __restrict__ filt,   // {fx,fy} pairs, 2*S floats
                          const float* __restrict__ bars0,
                          const float* __restrict__ bars1,
                          const float* __restrict__ pts,
                          float* __restrict__ out,
                          int S, int P, int R,
                          long long gBase0, long long gBase1,
                          long long gcBase0, long long gcBase1) {
    __shared__ float sh[2 * S_MAX];     // interleaved {fx,fy} pairs

    // ---- async stage: global (filt) -> LDS, 16 bytes per lane-op ----------
    {
        const int nvec = (2 * S + 3) >> 2;             // 16-byte chunks
        unsigned lds_base = (unsigned)(uintptr_t)(&sh[0]);
        for (int i = threadIdx.x; i < nvec; i += BLK_THREADS) {
            unsigned laddr = lds_base + 16u * (unsigned)i;
            uint64_t gaddr = (uint64_t)(uintptr_t)(filt + 4 * i);
            asm volatile("global_load_async_to_lds_b128 %0, %1, off"
                         :: "v"(laddr), "v"(gaddr) : "memory");
        }
        asm volatile("s_wait_asynccnt 0x0" ::: "memory");
    }
    __syncthreads();

    const int wave = threadIdx.x >> 5;
    const int lane = threadIdx.x & 31;
    const int p    = blockIdx.x * WAVES_PER_BLK + wave;
    if (p >= P) return;
    const int r = blockIdx.y;
    const int b = blockIdx.z;

    const float* bars = b ? bars1 : bars0;
    const float s   = bars[p * R + r] + GRID_RES_C;
    const float px  = pts[2 * p];
    const float py  = pts[2 * p + 1];
    const float psum  = px + py;
    const float lastx = fmaf(2.0f, s, px);   // shift[-1] == +L == 2
    const float lasty = fmaf(2.0f, s, py);

    float* __restrict__ growf =
        out + (b ? gBase1 : gBase0) +
        (long long)(r * P + p) * (long long)(2 * S);

    bool aLx = false, aUx = false, aLy = false, aUy = false;

    const float4* __restrict__ sh4 = (const float4*)sh;
    const int Spair = S & ~1;                 // even prefix (pairs of j)

    // Each lane owns j = jb, jb+1 (jb even): float4 LDS read, float4 store.
    for (int jb = lane * 2; jb < Spair; jb += 64) {
        const float4 f = sh4[jb >> 1];        // {fx[jb], fy[jb], fx[jb+1], fy[jb+1]}
        const float2 v0 = eval_cell(f.x, f.y, s, px, py, psum, lastx, lasty,
                                    aLx, aUx, aLy, aUy);
        const float2 v1 = eval_cell(f.z, f.w, s, px, py, psum, lastx, lasty,
                                    aLx, aUx, aLy, aUy);
        *(float4*)(growf + 2 * jb) = make_float4(v0.x, v0.y, v1.x, v1.y);
    }
    if ((S & 1) && lane == 0) {               // odd-S tail (not hit for S=6000)
        const int j = S - 1;
        const float2 v = eval_cell(sh[2 * j], sh[2 * j + 1], s, px, py,
                                   psum, lastx, lasty, aLx, aUx, aLy, aUy);
        *(float2*)(growf + 2 * j) = v;
    }

    // wave32 any-reductions for Gc
    const bool AUx = __ballot((int)aUx) != 0ull;
    const bool ALx = __ballot((int)aLx) != 0ull;
    const bool AUy = __ballot((int)aUy) != 0ull;
    const bool ALy = __ballot((int)aLy) != 0ull;

    if (lane == 0) {
        float* gc = out + (b ? gcBase1 : gcBase0) + 2LL * (long long)(r * P + p);
        gc[0] = AUx ? -1.0f : (ALx ? 1.0f : 0.0f);
        gc[1] = AUy ? -1.0f : (ALy ? 1.0f : 0.0f);
    }
}

// ---------------------------------------------------------------------------
extern "C" void kernel_launch(void* const* d_in, const int* in_sizes, int n_in,
                              void* d_out, int out_size, void* d_ws, size_t ws_size,
                              hipStream_t stream) {
    const float* f_v   = (const float*)d_in[0];
    const int*   edges = (const int*)  d_in[1];
    const float* bars0 = (const float*)d_in[2];
    const float* bars1 = (const float*)d_in[3];
    const float* pts   = (const float*)d_in[4];
    float* out = (float*)d_out;

    const int V = in_sizes[0] / 2;         // 1200
    const int E = in_sizes[1] / 2;         // 4800
    const int P = in_sizes[4] / 2;         // 2500
    const int R = in_sizes[2] / P;         // 3
    const int S = V + E;                   // 6000
    const int PR = P * R;

    float* filt = (float*)d_ws;            // interleaved {fx,fy} pairs, 2*S floats

    // Output layout (flat, return order): bars0 | bars1 | G0 | G1 | Gc0 | Gc1
    const long long gElems  = (long long)R * P * 2LL * S;   // per set
    const long long gBase0  = 2LL * PR;
    const long long gBase1  = gBase0 + gElems;
    const long long gcBase0 = gBase1 + gElems;
    const long long gcBase1 = gcBase0 + 2LL * PR;

    // Kernel A: filtration + bars copy
    {
        int n = PR;
        if (V > n) n = V;
        if (E > n) n = E;
        int blocks = (n + BLK_THREADS - 1) / BLK_THREADS;
        build_filtration_kernel<<<blocks, BLK_THREADS, 0, stream>>>(
            f_v, edges, bars0, bars1, out, filt, V, E, PR);
    }

    // Kernel B: G and Gc for both bar sets
    {
        dim3 grid((P + WAVES_PER_BLK - 1) / WAVES_PER_BLK, R, 2);
        grad_matrices_kernel<<<grid, BLK_THREADS, 0, stream>>>(
            filt, bars0, bars1, pts, out, S, P, R,
            gBase0, gBase1, gcBase0, gcBase1);
    }
}